// RegionFeatureInjection_1486058684825
// MI455X (gfx1250) — compile-verified
//
#include <hip/hip_runtime.h>
#include <hip/hip_bf16.h>

// ---------------------------------------------------------------------------
// RegionFeatureInjection for MI455X (gfx1250, wave32)
//   B=4, C=1280, H=64, W=64, N=16, DR=512, HM=WM=512
// ---------------------------------------------------------------------------

#define RFI_B   4
#define RFI_C   1280
#define RFI_H   64
#define RFI_W   64
#define RFI_N   16
#define RFI_DR  512
#define RFI_HM  512
#define RFI_WM  512

typedef float v2f __attribute__((ext_vector_type(2)));
typedef float v4f __attribute__((ext_vector_type(4)));
typedef float v8f __attribute__((ext_vector_type(8)));
typedef int   v4i __attribute__((ext_vector_type(4)));

// ---------------------------------------------------------------------------
// Kernel 1: proj(16,1280) = rf(16,512) @ Wp(1280,512)^T + bias, via
// V_WMMA_F32_16X16X4_F32. One wave (32 threads) per 16-column tile -> 80 waves.
//
// A fragment (16x4 f32, ISA 7.12.2): lanes 0-15 hold A[m, k..k+1] in v[0..1],
// lanes 16-31 hold A[m, k+2..k+3]. B (4x16) mirrors that with N striped across
// lanes: lanes 0-15 hold B[k..k+1, n], lanes 16-31 hold B[k+2..k+3, n].
// Since B = Wp^T, B[kk, n] = Wp[c0+n, kk] -> contiguous float2 along DR.
// ---------------------------------------------------------------------------
__global__ void __launch_bounds__(32)
rfi_proj_wmma(const float* __restrict__ rf,    // (16, 512)
              const float* __restrict__ Wp,    // (1280, 512)
              const float* __restrict__ bias,  // (1280)
              float* __restrict__ proj)        // (16, 1280)
{
    const int lane = threadIdx.x;     // 0..31, full wave, no divergence
    const int c0   = blockIdx.x * 16; // column tile base (0..1279)
    const int half = lane >> 4;       // 0: K={k,k+1}  1: K={k+2,k+3}
    const int l15  = lane & 15;

    const float* aRow = rf + (size_t)l15 * RFI_DR + half * 2;
    const float* bRow = Wp + (size_t)(c0 + l15) * RFI_DR + half * 2;

    v8f acc = {};
    #pragma unroll 8
    for (int k = 0; k < RFI_DR; k += 4) {
        v2f a = *(const v2f*)(aRow + k);   // global_load_b64
        v2f b = *(const v2f*)(bRow + k);   // global_load_b64
        // D = A x B + C  (8 args: neg_a, A, neg_b, B, c_mod, C, reuse_a, reuse_b)
        acc = __builtin_amdgcn_wmma_f32_16x16x4_f32(
            false, a, false, b, (short)0, acc, false, false);
    }

    const float bv = bias[c0 + l15];
    // C/D layout: VGPR v -> M = v + 8*half, N = lane&15
    #pragma unroll
    for (int v = 0; v < 8; ++v) {
        const int m = v + half * 8;
        proj[(size_t)m * RFI_C + c0 + l15] = acc[v] + bv;
    }
}

// ---------------------------------------------------------------------------
// Kernel 2: per-pixel last-writer region index.
// Nearest resize 512->64 is src = (dst*512)//64 = dst*8 -> stride-8 sampling.
// idx = last n with mask > 0.5, else -1.  4096 pixels total.
// ---------------------------------------------------------------------------
__global__ void __launch_bounds__(256)
rfi_build_idx(const float* __restrict__ masks,  // (16, 512, 512)
              int* __restrict__ idxmap)         // (64*64)
{
    const int p = blockIdx.x * blockDim.x + threadIdx.x;
    if (p >= RFI_H * RFI_W) return;
    const int h = p >> 6;
    const int w = p & 63;
    const float* base = masks + (size_t)(h * 8) * RFI_WM + (size_t)(w * 8);
    int idx = -1;
    #pragma unroll
    for (int n = 0; n < RFI_N; ++n) {
        const float m = base[(size_t)n * RFI_HM * RFI_WM];
        if (m > 0.5f) idx = n;
    }
    idxmap[p] = idx;
}

// ---------------------------------------------------------------------------
// Kernel 3: out[b,c,h,w] = spatial[b,c,h,w] + (idx[h,w]>=0 ? proj[idx,c] : 0)
// The 168 MB streaming pass: float4 per thread along W, nontemporal b128
// load/store (zero reuse), proj/idx gathers hit L0/L2 (96 KB total).
// Clamp-index + select keeps the gather unconditional (v_cndmask, no branch).
// ---------------------------------------------------------------------------
__global__ void __launch_bounds__(256)
rfi_inject_add(const float* __restrict__ spatial,
               const float* __restrict__ proj,
               const int*   __restrict__ idxmap,
               float* __restrict__ out,
               int total4)
{
    const int t = blockIdx.x * blockDim.x + threadIdx.x;
    if (t >= total4) return;

    const long long base = (long long)t * 4;
    const int w0 = (int)(base & 63);
    const int h  = (int)((base >> 6) & 63);
    const int c  = (int)((base >> 12) % RFI_C);
    const int p  = h * RFI_W + w0;                 // multiple of 4 -> v4i ok

    const v4i idx = *(const v4i*)(idxmap + p);
    const v4f sp  = __builtin_nontemporal_load((const v4f*)(spatial + base));

    const int ix = idx.x > 0 ? idx.x : 0;
    const int iy = idx.y > 0 ? idx.y : 0;
    const int iz = idx.z > 0 ? idx.z : 0;
    const int iw = idx.w > 0 ? idx.w : 0;

    const float fx = proj[(size_t)ix * RFI_C + c];
    const float fy = proj[(size_t)iy * RFI_C + c];
    const float fz = proj[(size_t)iz * RFI_C + c];
    const float fw = proj[(size_t)iw * RFI_C + c];

    v4f o;
    o.x = sp.x + (idx.x >= 0 ? fx : 0.0f);
    o.y = sp.y + (idx.y >= 0 ? fy : 0.0f);
    o.z = sp.z + (idx.z >= 0 ? fz : 0.0f);
    o.w = sp.w + (idx.w >= 0 ? fw : 0.0f);

    __builtin_nontemporal_store(o, (v4f*)(out + base));
}

// ---------------------------------------------------------------------------
// Launch
// ---------------------------------------------------------------------------
extern "C" void kernel_launch(void* const* d_in, const int* in_sizes, int n_in,
                              void* d_out, int out_size, void* d_ws, size_t ws_size,
                              hipStream_t stream) {
    const float* spatial = (const float*)d_in[0];  // (4,1280,64,64)
    const float* rf      = (const float*)d_in[1];  // (16,512)
    const float* masks   = (const float*)d_in[2];  // (16,512,512)
    const float* Wp      = (const float*)d_in[3];  // (1280,512)
    const float* bias    = (const float*)d_in[4];  // (1280)
    float* out           = (float*)d_out;          // (4,1280,64,64)

    // Workspace layout: proj (16*1280 f32 = 80 KB) | idxmap (4096 i32 = 16 KB)
    float* proj   = (float*)d_ws;
    int*   idxmap = (int*)((char*)d_ws + (size_t)RFI_N * RFI_C * sizeof(float));

    // K1: 80 column tiles, one wave each
    rfi_proj_wmma<<<RFI_C / 16, 32, 0, stream>>>(rf, Wp, bias, proj);

    // K2: 4096 pixels
    rfi_build_idx<<<(RFI_H * RFI_W + 255) / 256, 256, 0, stream>>>(masks, idxmap);

    // K3: streaming add, float4 per thread
    const int total4 = (RFI_B * RFI_C * RFI_H * RFI_W) / 4;   // 5,242,880
    rfi_inject_add<<<(total4 + 255) / 256, 256, 0, stream>>>(spatial, proj, idxmap, out, total4);
}